// LINK_47107201303128
// MI455X (gfx1250) — compile-verified
//
#include <hip/hip_runtime.h>

// ---------------------------------------------------------------------------
// LINK forward: logits = A @ W.T + b   (A = COO adjacency, E=3.2M, N=100K)
// Memory-bound SpMM: ~3.4 GB traffic -> ~150us floor @ 23.3 TB/s.
// Strategy:
//   1) transpose W [128,N] -> Wt [N,128] in d_ws  (Wt = 51.2MB, L2-resident)
//   2) init out with bias via v_wmma_f32_16x16x4_f32 broadcast (store-bound)
//   3) wave-per-edge gather(float4/lane) + global_atomic_add_f32 scatter,
//      with global_prefetch_b8 for the next edge's Wt row.
// ---------------------------------------------------------------------------

#define OUTC 128

typedef float v2f __attribute__((ext_vector_type(2)));
typedef float v8f __attribute__((ext_vector_type(8)));

// ---- 1) W [128, N] (row-major) -> Wt [N, 128], write-coalesced ------------
__global__ void transpose_w(const float* __restrict__ W, float* __restrict__ Wt,
                            int n_nodes) {
    size_t total  = (size_t)n_nodes * OUTC;
    size_t stride = (size_t)gridDim.x * blockDim.x;
    for (size_t i = (size_t)blockIdx.x * blockDim.x + threadIdx.x; i < total; i += stride) {
        size_t n = i >> 7;          // i / 128
        int    o = (int)(i & 127);  // i % 128
        Wt[i] = W[(size_t)o * n_nodes + n];
    }
}

// ---- 2) out[n][o] = bias[o], broadcast via WMMA (16x16 tiles, wave32) -----
__global__ void bias_init(const float* __restrict__ bias, float* __restrict__ out,
                          int n_nodes) {
    int lane = threadIdx.x & 31;
    long long wid    = (long long)blockIdx.x * (blockDim.x >> 5) + (threadIdx.x >> 5);
    long long ntiles = (long long)(n_nodes / 16) * (OUTC / 16);
    if (wid >= ntiles) return;                 // wave-uniform: EXEC stays all-1s
    int tm = (int)(wid / (OUTC / 16)) * 16;    // tile row (node)
    int tn = (int)(wid % (OUTC / 16)) * 16;    // tile col (channel)

    float bchunk = (lane < 16) ? bias[tn + lane] : 0.0f;

#if __has_builtin(__builtin_amdgcn_wmma_f32_16x16x4_f32)
    // A[16x4]: A[m][0]=1 else 0  (lanes 0-15: VGPR0=K0, VGPR1=K1; lanes 16-31: K2,K3)
    // B[4x16]: B[0][n]=bias[tn+n] else 0
    // => D[m][n] = bias[tn+n] for all 16 rows.
    v2f a, b;
    a.x = (lane < 16) ? 1.0f : 0.0f;  a.y = 0.0f;
    b.x = bchunk;                     b.y = 0.0f;
    v8f c = {};
    c = __builtin_amdgcn_wmma_f32_16x16x4_f32(false, a, false, b, (short)0, c,
                                              false, false);
    // C/D layout: VGPR r, lanes 0-15 -> (M=r, N=lane); lanes 16-31 -> (M=8+r, N=lane-16)
    int row0 = tm + ((lane < 16) ? 0 : 8);
    int col  = tn + (lane & 15);
#pragma unroll
    for (int r = 0; r < 8; ++r)
        out[(size_t)(row0 + r) * OUTC + col] = c[r];
#else
    // Fallback: plain stores (no WMMA available in this toolchain)
    int row0 = tm + ((lane < 16) ? 0 : 8);
    int col  = tn + (lane & 15);
    float v  = bias[col];
#pragma unroll
    for (int r = 0; r < 8; ++r)
        out[(size_t)(row0 + r) * OUTC + col] = v;
#endif
}

// tail rows when n_nodes % 16 != 0 (not hit for N=100000, kept for generality)
__global__ void bias_tail(const float* __restrict__ bias, float* __restrict__ out,
                          int n_nodes, int start) {
    int o = threadIdx.x;  // 128 threads
    float v = bias[o];
    for (int n = start + blockIdx.x; n < n_nodes; n += gridDim.x)
        out[(size_t)n * OUTC + o] = v;
}

// ---- 3) edge scatter: one wave per edge, float4 per lane ------------------
template <bool TRANSPOSED>
__global__ void edge_scatter(const int* __restrict__ erow, const int* __restrict__ ecol,
                             const float* __restrict__ Wsrc, float* __restrict__ out,
                             int n_edges, int n_nodes) {
    int lane = threadIdx.x & 31;
    long long wid = (long long)blockIdx.x * (blockDim.x >> 5) + (threadIdx.x >> 5);
    long long nw  = (long long)gridDim.x * (blockDim.x >> 5);

    for (long long e = wid; e < n_edges; e += nw) {
        int r = erow[e];
        int c = ecol[e];

        // Prefetch next iteration's gather row (lowers to global_prefetch_b8).
        long long en = e + nw;
        if (en < n_edges) {
            int cn = ecol[en];
            const float* p = TRANSPOSED
                ? (Wsrc + (size_t)cn * OUTC + lane * 4)
                : (Wsrc + (size_t)(lane * 4) * n_nodes + cn);
            __builtin_prefetch(p, 0, 1);
        }

        float4 v;
        if (TRANSPOSED) {
            // contiguous 512B per edge, 16B per lane
            v = ((const float4*)(Wsrc + (size_t)c * OUTC))[lane];
        } else {
            size_t base = (size_t)(lane * 4) * n_nodes + c;
            v.x = Wsrc[base];
            v.y = Wsrc[base + (size_t)n_nodes];
            v.z = Wsrc[base + 2 * (size_t)n_nodes];
            v.w = Wsrc[base + 3 * (size_t)n_nodes];
        }

        float* o = out + (size_t)r * OUTC + lane * 4;
        atomicAdd(o + 0, v.x);
        atomicAdd(o + 1, v.y);
        atomicAdd(o + 2, v.z);
        atomicAdd(o + 3, v.w);
    }
}

// ---------------------------------------------------------------------------
extern "C" void kernel_launch(void* const* d_in, const int* in_sizes, int n_in,
                              void* d_out, int out_size, void* d_ws, size_t ws_size,
                              hipStream_t stream) {
    // inputs: 0=x (unused), 1=edge_index [2,E], 2=W_weight [128,N], 3=W_bias [128]
    const int*   edge = (const int*)d_in[1];
    const float* Ww   = (const float*)d_in[2];
    const float* Wb   = (const float*)d_in[3];
    float*       out  = (float*)d_out;

    int E       = in_sizes[1] / 2;
    int n_nodes = out_size / OUTC;
    const int* erow = edge;
    const int* ecol = edge + E;

    size_t need = (size_t)n_nodes * OUTC * sizeof(float);
    bool   useT = (ws_size >= need);
    float* Wt   = (float*)d_ws;

    // 1) transpose (only if workspace is big enough)
    if (useT) {
        long long total  = (long long)n_nodes * OUTC;
        int       blocks = (int)((total + 255) / 256);
        transpose_w<<<blocks, 256, 0, stream>>>(Ww, Wt, n_nodes);
    }

    // 2) bias init (WMMA broadcast)
    {
        long long ntiles = (long long)(n_nodes / 16) * (OUTC / 16);
        int blocks = (int)((ntiles + 7) / 8);  // 8 waves / 256-thread block
        if (blocks > 0)
            bias_init<<<blocks, 256, 0, stream>>>(Wb, out, n_nodes);
        if (n_nodes % 16)
            bias_tail<<<64, OUTC, 0, stream>>>(Wb, out, n_nodes, (n_nodes / 16) * 16);
    }

    // 3) scatter-add over edges
    {
        int blocks = 4096;  // 32768 waves, ~98 edges each -> prefetch pipelines
        if (useT)
            edge_scatter<true><<<blocks, 256, 0, stream>>>(erow, ecol, Wt, out, E, n_nodes);
        else
            edge_scatter<false><<<blocks, 256, 0, stream>>>(erow, ecol, Ww, out, E, n_nodes);
    }
}